// NetlistGNN_48369921688187
// MI455X (gfx1250) — compile-verified
//
#include <hip/hip_runtime.h>
#include <cstddef>

// ---------------------------------------------------------------------------
// NetlistGNN forward for MI455X (gfx1250, wave32).
// Dense [rows,K]@[K,NCOLS] projections run on V_WMMA_F32_16X16X4_F32 (full
// fp32, 16x16 tile per wave, compile-time K/NCOLS -> fully unrolled, branch-
// free inner loop). Sparse segment ops run as L2-resident atomics (float
// atomicAdd for sums; int atomicMax on nonnegative float bits for max).
// ---------------------------------------------------------------------------

#define NN_ 100000
#define NT_ 30000
#define NP_ 400000
#define NE_ 1000000
#define L_ 3

typedef float v2f __attribute__((ext_vector_type(2)));
typedef float v8f __attribute__((ext_vector_type(8)));

#if defined(__HIP_DEVICE_COMPILE__) && !__has_builtin(__builtin_amdgcn_wmma_f32_16x16x4_f32)
#error "gfx1250 f32 WMMA builtin unavailable"
#endif

__device__ __forceinline__ float d_lrelu(float x) { return x > 0.f ? x : 0.01f * x; }
__device__ __forceinline__ float d_ssp(float x) {
  // softplus(x) - log(2), numerically stable
  return fmaxf(x, 0.f) + log1pf(__expf(-fabsf(x))) - 0.6931471805599453f;
}
__device__ __forceinline__ float d_sigm(float x) { return 1.f / (1.f + __expf(-x)); }

template <int ACT> __device__ __forceinline__ float d_act(float x) {
  if (ACT == 1) return d_lrelu(x);
  if (ACT == 2) return d_ssp(x);
  if (ACT == 3) return fmaxf(x, 0.f);
  return x;
}

// ---------------------------------------------------------------------------
// WMMA GEMM: Y[rows,NCOLS] = act( (rowscale.*X1)@W1 + X2@W2 + bias ), with
// optional elementwise max against `maxwith` before activation.
// NCOLS multiple of 16; K1,K2 multiples of 4 (compile-time).
// 8 waves/block, one 16x16 tile per wave. Row-edge handling is branchless:
// clamp the row index and scale A by a 0/1 mask so EXEC stays full for WMMA.
// ---------------------------------------------------------------------------
template <int ACT, int NCOLS, int K1, int K2>
__global__ __launch_bounds__(256) void gemm_wmma_f32(
    const float* __restrict__ X1, const float* __restrict__ W1,
    const float* __restrict__ X2, const float* __restrict__ W2,
    const float* __restrict__ bias, const float* __restrict__ rowscale,
    const float* __restrict__ maxwith, float* __restrict__ Y, int rows) {
  const int lane = threadIdx.x & 31;
  const int wave = threadIdx.x >> 5;
  constexpr int coltiles = NCOLS / 16;
  const long tile = (long)blockIdx.x * 8 + wave;
  const int rtile = (int)(tile / coltiles);
  const int ctile = (int)(tile % coltiles);
  const int rowbase = rtile * 16;
  if (rowbase >= rows) return;  // uniform per wave: EXEC stays all-ones below

  const int half = lane >> 4;   // which K/M half this lane serves
  const int n = lane & 15;      // column within tile (B/C/D), row within tile (A)
  const int cb = ctile * 16;

  const int m = rowbase + n;
  const bool mok = m < rows;
  const int mc = mok ? m : rows - 1;     // clamped (always in-bounds) A row
  float fm = mok ? 1.f : 0.f;            // validity mask, folded with rowscale
  if (rowscale) fm *= rowscale[mc];

  v8f acc = {0.f, 0.f, 0.f, 0.f, 0.f, 0.f, 0.f, 0.f};
  {
    const float* xr = X1 + (size_t)mc * K1 + 2 * half;   // 8B-aligned (even idx)
    const float* wp = W1 + (size_t)(2 * half) * NCOLS + cb + n;
#pragma unroll
    for (int k0 = 0; k0 < K1; k0 += 4) {
      v2f a = *(const v2f*)(xr + k0);
      a.x *= fm;
      a.y *= fm;
      v2f b;
      b.x = wp[(size_t)k0 * NCOLS];
      b.y = wp[(size_t)(k0 + 1) * NCOLS];
      acc = __builtin_amdgcn_wmma_f32_16x16x4_f32(false, a, false, b, (short)0, acc,
                                                  false, false);
    }
  }
  if (K2 > 0) {
    const float fm2 = mok ? 1.f : 0.f;
    const float* xr = X2 + (size_t)mc * K2 + 2 * half;
    const float* wp = W2 + (size_t)(2 * half) * NCOLS + cb + n;
#pragma unroll
    for (int k0 = 0; k0 < K2; k0 += 4) {
      v2f a = *(const v2f*)(xr + k0);
      a.x *= fm2;
      a.y *= fm2;
      v2f b;
      b.x = wp[(size_t)k0 * NCOLS];
      b.y = wp[(size_t)(k0 + 1) * NCOLS];
      acc = __builtin_amdgcn_wmma_f32_16x16x4_f32(false, a, false, b, (short)0, acc,
                                                  false, false);
    }
  }

  const float bv = bias ? bias[cb + n] : 0.f;
#pragma unroll
  for (int j = 0; j < 8; ++j) {
    const int mr = rowbase + j + 8 * half;
    if (mr < rows) {
      float v = acc[j] + bv;
      if (maxwith) v = fmaxf(v, maxwith[(size_t)mr * NCOLS + cb + n]);
      Y[(size_t)mr * NCOLS + cb + n] = d_act<ACT>(v);
    }
  }
}

// ---------------------------------------------------------------------------
// Scalar / sparse kernels
// ---------------------------------------------------------------------------
__global__ void k_edge_lin(const float* __restrict__ X, const float* __restrict__ W,
                           const float* __restrict__ b, float* __restrict__ Y, int n) {
  int e = blockIdx.x * blockDim.x + threadIdx.x;
  if (e >= n) return;
  float x0 = X[e * 4 + 0], x1 = X[e * 4 + 1], x2 = X[e * 4 + 2], x3 = X[e * 4 + 3];
#pragma unroll
  for (int j = 0; j < 8; ++j) {
    float v = x0 * W[j] + x1 * W[8 + j] + x2 * W[16 + j] + x3 * W[24 + j] + b[j];
    Y[(size_t)e * 8 + j] = d_lrelu(v);
  }
}

__global__ void k_deg(const int* __restrict__ src, const int* __restrict__ dst,
                      float* __restrict__ dsrc, float* __restrict__ ddst, int np) {
  int i = blockIdx.x * blockDim.x + threadIdx.x;
  if (i >= np) return;
  atomicAdd(&dsrc[src[i]], 1.f);
  atomicAdd(&ddst[dst[i]], 1.f);
}

__global__ void k_norm(float* __restrict__ d, int n) {
  int i = blockIdx.x * blockDim.x + threadIdx.x;
  if (i < n) d[i] = rsqrtf(fmaxf(d[i], 1.f));
}

// agg[dst] += node_feat[src] * src_norm[src]; 2 channels per thread.
__global__ void k_scatter_pins(const int* __restrict__ ps, const int* __restrict__ pd,
                               const float* __restrict__ nf, const float* __restrict__ sn,
                               float* __restrict__ agg, long total) {
  long i = (long)blockIdx.x * blockDim.x + threadIdx.x;
  if (i >= total) return;
  int p = (int)(i >> 5), h = (int)((i & 31) << 1);
  int s = ps[p], d = pd[p];
  float sv = sn[s];
  float2 v = *(const float2*)(nf + (size_t)s * 64 + h);
  atomicAdd(&agg[(size_t)d * 64 + h], v.x * sv);
  atomicAdd(&agg[(size_t)d * 64 + h + 1], v.y * sv);
}

// hcf[src] += hv[dst] * he[pin]; 2 channels per thread.
__global__ void k_scatter_cf(const int* __restrict__ ps, const int* __restrict__ pd,
                             const float* __restrict__ hv, const float* __restrict__ he,
                             float* __restrict__ hcf, long total) {
  long i = (long)blockIdx.x * blockDim.x + threadIdx.x;
  if (i >= total) return;
  int p = (int)(i >> 5), h = (int)((i & 31) << 1);
  int s = ps[p], d = pd[p];
  float2 a = *(const float2*)(hv + (size_t)d * 64 + h);
  float2 b = *(const float2*)(he + (size_t)p * 64 + h);
  atomicAdd(&hcf[(size_t)s * 64 + h], a.x * b.x);
  atomicAdd(&hcf[(size_t)s * 64 + h + 1], a.y * b.y);
}

__global__ void k_ew(const float* __restrict__ ef, const float* __restrict__ gw,
                     const float* __restrict__ gb, float* __restrict__ ew, int n) {
  int e = blockIdx.x * blockDim.x + threadIdx.x;
  if (e >= n) return;
  float a = gb[0];
#pragma unroll
  for (int j = 0; j < 8; ++j) a += ef[(size_t)e * 8 + j] * gw[j];
  ew[e] = d_sigm(a);
}

// hneigh[dst] = max(hneigh[dst], hp[src]*ew); messages >= 0, so integer
// atomicMax on the raw bits is exact (buffer zero-initialized). 2 ch/thread.
__global__ void k_scatter_max(const int* __restrict__ es, const int* __restrict__ ed,
                              const float* __restrict__ hp, const float* __restrict__ ew,
                              int* __restrict__ hng, long total) {
  long i = (long)blockIdx.x * blockDim.x + threadIdx.x;
  if (i >= total) return;
  int e = (int)(i >> 5), h = (int)((i & 31) << 1);
  float w = ew[e];
  float2 v = *(const float2*)(hp + (size_t)es[e] * 64 + h);
  int* base = hng + (size_t)ed[e] * 64 + h;
  atomicMax(base, __float_as_int(v.x * w));
  atomicMax(base + 1, __float_as_int(v.y * w));
}

// sigmoid( H[rows,64] @ W[64,T] + b )
__global__ void k_head(const float* __restrict__ H, const float* __restrict__ W,
                       const float* __restrict__ b, float* __restrict__ out, int rows, int T) {
  long i = (long)blockIdx.x * blockDim.x + threadIdx.x;
  if (i >= (long)rows * T) return;
  int r = (int)(i / T), t = (int)(i % T);
  float a = b[t];
  const float* hr = H + (size_t)r * 64;
#pragma unroll 8
  for (int k = 0; k < 64; ++k) a += hr[k] * W[k * T + t];
  out[i] = d_sigm(a);
}

// ---------------------------------------------------------------------------
// Host side
// ---------------------------------------------------------------------------
static inline int cdiv(long a, int b) { return (int)((a + b - 1) / b); }

template <int ACT, int NCOLS, int K1, int K2>
static inline void launch_gemm(const float* X1, const float* W1, const float* X2,
                               const float* W2, const float* bias, const float* rowscale,
                               const float* maxwith, float* Y, int rows, hipStream_t s) {
  long tiles = (long)((rows + 15) / 16) * (NCOLS / 16);
  gemm_wmma_f32<ACT, NCOLS, K1, K2><<<dim3((unsigned)((tiles + 7) / 8)), dim3(256), 0, s>>>(
      X1, W1, X2, W2, bias, rowscale, maxwith, Y, rows);
}

extern "C" void kernel_launch(void* const* d_in, const int* in_sizes, int n_in,
                              void* d_out, int out_size, void* d_ws, size_t ws_size,
                              hipStream_t stream) {
  (void)in_sizes; (void)n_in; (void)out_size; (void)ws_size;
  const float* in_node = (const float*)d_in[0];
  const float* in_net  = (const float*)d_in[1];
  const float* in_pin  = (const float*)d_in[2];
  const float* in_edge = (const float*)d_in[3];
  const int* pins_src  = (const int*)d_in[4];
  const int* pins_dst  = (const int*)d_in[5];
  const int* near_src  = (const int*)d_in[6];
  const int* near_dst  = (const int*)d_in[7];
  const float* node_lin_W = (const float*)d_in[8];  const float* node_lin_b = (const float*)d_in[9];
  const float* net_lin_W  = (const float*)d_in[10]; const float* net_lin_b  = (const float*)d_in[11];
  const float* pin_lin_W  = (const float*)d_in[12]; const float* pin_lin_b  = (const float*)d_in[13];
  const float* edge_lin_W = (const float*)d_in[14]; const float* edge_lin_b = (const float*)d_in[15];
  const float* geom_W = (const float*)d_in[16]; const float* geom_b = (const float*)d_in[17];
  const float* pins_W = (const float*)d_in[18]; const float* pins_b = (const float*)d_in[19];
  const float* cf_node_W = (const float*)d_in[20]; const float* cf_node_b = (const float*)d_in[21];
  const float* cf_e1_W   = (const float*)d_in[22]; const float* cf_e1_b   = (const float*)d_in[23];
  const float* cf_e2_W   = (const float*)d_in[24]; const float* cf_e2_b   = (const float*)d_in[25];
  const float* cf_out_W  = (const float*)d_in[26]; const float* cf_out_b  = (const float*)d_in[27];
  const float* sage_pool_W = (const float*)d_in[28]; const float* sage_pool_b = (const float*)d_in[29];
  const float* sage_self_W  = (const float*)d_in[30];
  const float* sage_neigh_W = (const float*)d_in[31];
  const float* sage_bias    = (const float*)d_in[32];
  const float* out1_W = (const float*)d_in[33]; const float* out1_b = (const float*)d_in[34];
  const float* out2_W = (const float*)d_in[35]; const float* out2_b = (const float*)d_in[36];
  const float* out3_W = (const float*)d_in[37]; const float* out3_b = (const float*)d_in[38];
  const float* outn1_W = (const float*)d_in[39]; const float* outn1_b = (const float*)d_in[40];
  const float* outn2_W = (const float*)d_in[41]; const float* outn2_b = (const float*)d_in[42];
  const float* outn3_W = (const float*)d_in[43]; const float* outn3_b = (const float*)d_in[44];

  float* ws = (float*)d_ws;
  size_t off = 0;
  auto alloc = [&](size_t nelem) { float* p = ws + off; off += nelem; return p; };
  float* node_a = alloc((size_t)NN_ * 64);
  float* node_b = alloc((size_t)NN_ * 64);
  float* net_a  = alloc((size_t)NT_ * 64);
  float* net_b  = alloc((size_t)NT_ * 64);
  float* pin_f  = alloc((size_t)NP_ * 16);
  float* edge_f = alloc((size_t)NE_ * 8);
  float* src_nrm = alloc(NN_);
  float* dst_nrm = alloc(NT_);
  float* ew   = alloc(NE_);
  float* agg  = alloc((size_t)NT_ * 64);
  float* hv   = alloc((size_t)NT_ * 64);
  float* he1  = alloc((size_t)NP_ * 64);
  float* he2  = alloc((size_t)NP_ * 64);
  float* hcf  = alloc((size_t)NN_ * 64);
  float* ncf  = alloc((size_t)NN_ * 64);
  float* hp   = alloc((size_t)NN_ * 64);
  float* hng  = alloc((size_t)NN_ * 64);

  const int TB = 256;

  // ---- input projections ----
  launch_gemm<1, 64, 16, 0>(in_node, node_lin_W, nullptr, nullptr, node_lin_b, nullptr, nullptr, node_a, NN_, stream);
  launch_gemm<1, 64, 8, 0>(in_net, net_lin_W, nullptr, nullptr, net_lin_b, nullptr, nullptr, net_a, NT_, stream);
  launch_gemm<1, 16, 8, 0>(in_pin, pin_lin_W, nullptr, nullptr, pin_lin_b, nullptr, nullptr, pin_f, NP_, stream);
  k_edge_lin<<<cdiv(NE_, TB), TB, 0, stream>>>(in_edge, edge_lin_W, edge_lin_b, edge_f, NE_);

  // ---- degree norms ----
  (void)hipMemsetAsync(src_nrm, 0, (size_t)NN_ * 4, stream);
  (void)hipMemsetAsync(dst_nrm, 0, (size_t)NT_ * 4, stream);
  k_deg<<<cdiv(NP_, TB), TB, 0, stream>>>(pins_src, pins_dst, src_nrm, dst_nrm, NP_);
  k_norm<<<cdiv(NN_, TB), TB, 0, stream>>>(src_nrm, NN_);
  k_norm<<<cdiv(NT_, TB), TB, 0, stream>>>(dst_nrm, NT_);

  float* node_f = node_a; float* node_n = node_b;
  float* net_f = net_a;   float* net_n = net_b;

  for (int i = 0; i < L_; ++i) {
    const size_t w64 = (size_t)i * 64 * 64, b64 = (size_t)i * 64;
    // --- 'pins' GraphConv (node -> net), norm='both'; fuse lrelu (new net_feat)
    (void)hipMemsetAsync(agg, 0, (size_t)NT_ * 64 * 4, stream);
    k_scatter_pins<<<cdiv((long)NP_ * 32, TB), TB, 0, stream>>>(pins_src, pins_dst, node_f, src_nrm, agg, (long)NP_ * 32);
    launch_gemm<1, 64, 64, 0>(agg, pins_W + w64, nullptr, nullptr, pins_b + b64, dst_nrm, nullptr, net_n, NT_, stream);

    // --- 'pinned' CFConv (net -> node)
    launch_gemm<0, 64, 64, 0>(net_f, cf_node_W + w64, nullptr, nullptr, cf_node_b + b64, nullptr, nullptr, hv, NT_, stream);
    launch_gemm<2, 64, 16, 0>(pin_f, cf_e1_W + (size_t)i * 16 * 64, nullptr, nullptr, cf_e1_b + b64, nullptr, nullptr, he1, NP_, stream);
    launch_gemm<2, 64, 64, 0>(he1, cf_e2_W + w64, nullptr, nullptr, cf_e2_b + b64, nullptr, nullptr, he2, NP_, stream);
    (void)hipMemsetAsync(hcf, 0, (size_t)NN_ * 64 * 4, stream);
    k_scatter_cf<<<cdiv((long)NP_ * 32, TB), TB, 0, stream>>>(pins_src, pins_dst, hv, he2, hcf, (long)NP_ * 32);
    launch_gemm<2, 64, 64, 0>(hcf, cf_out_W + w64, nullptr, nullptr, cf_out_b + b64, nullptr, nullptr, ncf, NN_, stream);

    // --- 'near' SAGEConv(pool) with sigmoid edge gates
    k_ew<<<cdiv(NE_, TB), TB, 0, stream>>>(edge_f, geom_W + (size_t)i * 8, geom_b + i, ew, NE_);
    launch_gemm<3, 64, 64, 0>(node_f, sage_pool_W + w64, nullptr, nullptr, sage_pool_b + b64, nullptr, nullptr, hp, NN_, stream);
    (void)hipMemsetAsync(hng, 0, (size_t)NN_ * 64 * 4, stream);
    k_scatter_max<<<cdiv((long)NE_ * 32, TB), TB, 0, stream>>>(near_src, near_dst, hp, ew, (int*)hng, (long)NE_ * 32);

    // node_new = lrelu( max( node_cf, self@W + neigh@W + bias ) )  (fused)
    launch_gemm<1, 64, 64, 64>(node_f, sage_self_W + w64, hng, sage_neigh_W + w64, sage_bias + b64, nullptr, ncf, node_n, NN_, stream);

    float* t;
    t = node_f; node_f = node_n; node_n = t;
    t = net_f;  net_f = net_n;   net_n = t;
  }

  // ---- output heads (concat handled as dual-K GEMM over stacked weight rows)
  float* h1 = hcf; float* h2 = ncf; float* g1 = agg; float* g2 = hv;
  float* outp = (float*)d_out;
  launch_gemm<1, 64, 16, 64>(in_node, out1_W, node_f, out1_W + 16 * 64, out1_b, nullptr, nullptr, h1, NN_, stream);
  launch_gemm<1, 64, 64, 0>(h1, out2_W, nullptr, nullptr, out2_b, nullptr, nullptr, h2, NN_, stream);
  k_head<<<cdiv((long)NN_ * 4, TB), TB, 0, stream>>>(h2, out3_W, out3_b, outp, NN_, 4);
  launch_gemm<1, 64, 8, 64>(in_net, outn1_W, net_f, outn1_W + 8 * 64, outn1_b, nullptr, nullptr, g1, NT_, stream);
  launch_gemm<1, 64, 64, 0>(g1, outn2_W, nullptr, nullptr, outn2_b, nullptr, nullptr, g2, NT_, stream);
  k_head<<<cdiv((long)NT_ * 1, TB), TB, 0, stream>>>(g2, outn3_W, outn3_b, outp + (size_t)NN_ * 4, NT_, 1);
}